// RecurrentGCN_62852551409688
// MI455X (gfx1250) — compile-verified
//
#include <hip/hip_runtime.h>
#include <math.h>

typedef float v2f __attribute__((ext_vector_type(2)));
typedef float v8f __attribute__((ext_vector_type(8)));

#define N_NODES 100000
#define N_EDGES 1600000
#define F_IN 35
#define F_OUT 5
#define HID 100
#define C_XH 40

#define GEMM_KMAX 40
#define GEMM_NMAX 112
#define ALDS_STRIDE 41

// ---------------- utility ----------------
__global__ void zero_kernel(float* __restrict__ p, long n) {
  long i = (long)blockIdx.x * blockDim.x + threadIdx.x;
  long stride = (long)gridDim.x * blockDim.x;
  for (; i < n; i += stride) p[i] = 0.0f;
}

// ---------------- degrees & edge normalization ----------------
__global__ void degree_kernel(const int* __restrict__ ei, const float* __restrict__ ew,
                              float* __restrict__ deg_out, float* __restrict__ deg_in) {
  int e = blockIdx.x * blockDim.x + threadIdx.x;
  if (e >= N_EDGES) return;
  int s = ei[e];
  int d = ei[N_EDGES + e];
  float w = ew[e];
  atomicAdd(&deg_out[s], w);
  atomicAdd(&deg_in[d], w);
}

__global__ void edgew_kernel(const int* __restrict__ ei, const float* __restrict__ ew,
                             const float* __restrict__ deg_out, const float* __restrict__ deg_in,
                             float* __restrict__ w_fwd, float* __restrict__ w_bwd,
                             float* __restrict__ outA) {
  int e = blockIdx.x * blockDim.x + threadIdx.x;
  if (e >= N_EDGES) return;
  int s = ei[e];
  int d = ei[N_EDGES + e];
  float w = ew[e];
  float wf = w / fmaxf(deg_out[s], 1e-12f);
  float wb = w / fmaxf(deg_in[d], 1e-12f);
  w_fwd[e] = wf;
  w_bwd[e] = wb;
  outA[e] = wf;   // second output "A" = normalized forward weights
}

// ---------------- generic f32 WMMA GEMM: Out[M,Cout] = In[M,Cin] @ W[Cin,Cout] ----------------
// 8 waves / block; one wave per 16-row tile. A and W staged in LDS (zero-padded) so the
// WMMA inner loop is branchless ds_load + v_wmma only.
__global__ void __launch_bounds__(256) gemm_wmma_kernel(const float* __restrict__ In,
                                                        const float* __restrict__ W,
                                                        float* __restrict__ Out, int M, int Cin,
                                                        int Cout) {
  __shared__ float Wlds[GEMM_KMAX * GEMM_NMAX];       // padded [Kp][Np], stride GEMM_NMAX
  __shared__ float Alds[8 * 16 * ALDS_STRIDE];        // per-wave 16 x Kp tile, stride 41
  int tid = threadIdx.x;
  int wave = tid >> 5;
  int lane = tid & 31;
  int hi = lane >> 4;   // 0: lanes 0-15, 1: lanes 16-31
  int l15 = lane & 15;
  int kchunks = (Cin + 3) >> 2;
  int ntiles = (Cout + 15) >> 4;
  int Kp = kchunks * 4;
  int Np = ntiles * 16;
  int row0 = (blockIdx.x * 8 + wave) * 16;

  // --- stage W cooperatively (branchless clamped loads + select) ---
  for (int idx = tid; idx < Kp * Np; idx += 256) {
    int k = idx / Np;
    int n = idx - k * Np;
    int ks = k < Cin ? k : Cin - 1;
    int ns = n < Cout ? n : Cout - 1;
    float v = W[ks * Cout + ns];
    Wlds[k * GEMM_NMAX + n] = (k < Cin && n < Cout) ? v : 0.0f;
  }
  // --- stage this wave's 16 x Kp A tile, coalesced reads ---
  float* Aw = Alds + wave * (16 * ALDS_STRIDE);
  for (int r = 0; r < 16; ++r) {
    int row = row0 + r;
    int rs = row < M ? row : M - 1;
    const float* ip = In + (long)rs * Cin;
    for (int c = lane; c < Kp; c += 32) {
      float v = ip[c < Cin ? c : 0];
      Aw[r * ALDS_STRIDE + c] = (row < M && c < Cin) ? v : 0.0f;
    }
  }
  __syncthreads();
  if (row0 >= M) return;   // wave-uniform; after barrier so no divergent sync

  const float* Al = Aw + l15 * ALDS_STRIDE;
  for (int nt = 0; nt < ntiles; ++nt) {
    int n = nt * 16 + l15;
    v8f c = (v8f)0.0f;
    for (int kc = 0; kc < kchunks; ++kc) {
      int kb = kc * 4 + hi * 2;  // ISA 16x4 A layout: lanes0-15 K=0,1; lanes16-31 K=2,3
      v2f a, b;
      a.x = Al[kb];
      a.y = Al[kb + 1];
      b.x = Wlds[kb * GEMM_NMAX + n];          // B 4x16: VGPR0 = K=0 / K=2 rows
      b.y = Wlds[(kb + 1) * GEMM_NMAX + n];    //         VGPR1 = K=1 / K=3 rows
      c = __builtin_amdgcn_wmma_f32_16x16x4_f32(false, a, false, b, (short)0, c, false, false);
    }
    // D layout: VGPR r -> M = r + hi*8, N = l15
#pragma unroll
    for (int r = 0; r < 8; ++r) {
      int row = row0 + r + hi * 8;
      if (row < M && n < Cout) Out[(long)row * Cout + n] = c[r];
    }
  }
}

// ---------------- MLP epilogue: h1 = relu(G1+b1)@W2+b2 ; build XH = [x | h1] ----------------
__global__ void mlp2_kernel(const float* __restrict__ x, const float* __restrict__ G1,
                            const float* __restrict__ b1, const float* __restrict__ W2,
                            const float* __restrict__ b2, float* __restrict__ h1,
                            float* __restrict__ XH) {
  __shared__ float W2s[HID * F_OUT];
  __shared__ float b1s[HID];
  for (int i = threadIdx.x; i < HID * F_OUT; i += blockDim.x) W2s[i] = W2[i];
  for (int i = threadIdx.x; i < HID; i += blockDim.x) b1s[i] = b1[i];
  __syncthreads();
  int n = blockIdx.x * blockDim.x + threadIdx.x;
  if (n >= N_NODES) return;
  float acc[F_OUT];
#pragma unroll
  for (int c = 0; c < F_OUT; ++c) acc[c] = b2[c];
  const float* g = G1 + (long)n * HID;
#pragma unroll 4
  for (int j = 0; j < HID; ++j) {
    float t = g[j] + b1s[j];
    t = t > 0.0f ? t : 0.0f;
#pragma unroll
    for (int c = 0; c < F_OUT; ++c) acc[c] += t * W2s[j * F_OUT + c];
  }
  float* xr = XH + (long)n * C_XH;
  const float* xs = x + (long)n * F_IN;
  for (int j = 0; j < F_IN; ++j) xr[j] = xs[j];
#pragma unroll
  for (int c = 0; c < F_OUT; ++c) {
    h1[(long)n * F_OUT + c] = acc[c];
    xr[F_IN + c] = acc[c];
  }
}

// ---------------- fold gate weights into combined projection matrices ----------------
// Wc_zr[40,50] cols: 0-4 Az, 5-9 Ar, 10-14 Fz1, 15-19 Fr1, 20-24 Fz2, 25-29 Fr2,
//                    30-34 Bz1, 35-39 Br1, 40-44 Bz2, 45-49 Br2
// Wc_h[40,25]  cols: 0-4 Ah, 5-9 Fh1, 10-14 Fh2, 15-19 Bh1, 20-24 Bh2
__global__ void buildw_kernel(const float* __restrict__ Wz, const float* __restrict__ Wr,
                              const float* __restrict__ Wh, float* __restrict__ Wc_zr,
                              float* __restrict__ Wc_h) {
  int idx = blockIdx.x * blockDim.x + threadIdx.x;
  auto W = [](const float* p, int d, int k, int i, int o) {
    return p[((d * 3 + k) * C_XH + i) * F_OUT + o];
  };
  if (idx < C_XH * 50) {
    int i = idx / 50, j = idx % 50;
    int b = j / 5, o = j % 5;
    const float* P = ((b & 1) == 0) ? Wz : Wr;
    float v;
    switch (b >> 1) {
      case 0: v = W(P, 0, 0, i, o) + W(P, 1, 0, i, o) - W(P, 0, 2, i, o) - W(P, 1, 2, i, o); break;
      case 1: v = W(P, 0, 1, i, o); break;
      case 2: v = W(P, 0, 2, i, o); break;
      case 3: v = W(P, 1, 1, i, o); break;
      default: v = W(P, 1, 2, i, o); break;
    }
    Wc_zr[idx] = v;
  }
  int idx2 = idx - C_XH * 50;
  if (idx2 >= 0 && idx2 < C_XH * 25) {
    int i = idx2 / 25, j = idx2 % 25;
    int b = j / 5, o = j % 5;
    float v;
    switch (b) {
      case 0: v = W(Wh, 0, 0, i, o) + W(Wh, 1, 0, i, o) - W(Wh, 0, 2, i, o) - W(Wh, 1, 2, i, o); break;
      case 1: v = W(Wh, 0, 1, i, o); break;
      case 2: v = W(Wh, 0, 2, i, o); break;
      case 3: v = W(Wh, 1, 1, i, o); break;
      default: v = W(Wh, 1, 2, i, o); break;
    }
    Wc_h[idx2] = v;
  }
}

// ---------------- sparse propagation: Out[sidx[e]] += w[e] * In[gidx[e]] (C channels) -------
__global__ void prop_kernel(const float* __restrict__ In, int in_stride, int in_off,
                            float* __restrict__ Out, int C, const int* __restrict__ gidx,
                            const int* __restrict__ sidx, const float* __restrict__ w) {
  int e = blockIdx.x * blockDim.x + threadIdx.x;
  if (e >= N_EDGES) return;
  int g = gidx[e];
  int s = sidx[e];
  float wt = w[e];
  const float* src = In + (long)g * in_stride + in_off;
  float* dst = Out + (long)s * C;
  for (int c = 0; c < C; ++c) atomicAdd(&dst[c], wt * src[c]);
}

// ---------------- Z/R gate eval; rewrite XH -> XHr in place ----------------
__global__ void gates_kernel(const float* __restrict__ Proj, const float* __restrict__ Qf,
                             const float* __restrict__ Qf2, const float* __restrict__ Qb,
                             const float* __restrict__ Qb2, const float* __restrict__ bz,
                             const float* __restrict__ br, const float* __restrict__ h1,
                             float* __restrict__ Z, float* __restrict__ XH) {
  int n = blockIdx.x * blockDim.x + threadIdx.x;
  if (n >= N_NODES) return;
  const float* p = Proj + (long)n * 50;
  const float* qf = Qf + (long)n * 20;
  const float* qf2 = Qf2 + (long)n * 10;
  const float* qb = Qb + (long)n * 20;
  const float* qb2 = Qb2 + (long)n * 10;
#pragma unroll
  for (int c = 0; c < F_OUT; ++c) {
    float zp = p[c] + bz[c] + qf[c] + 2.0f * qf2[c] + qb[c] + 2.0f * qb2[c];
    float rp = p[5 + c] + br[c] + qf[5 + c] + 2.0f * qf2[5 + c] + qb[5 + c] + 2.0f * qb2[5 + c];
    float z = 1.0f / (1.0f + expf(-zp));
    float r = 1.0f / (1.0f + expf(-rp));
    Z[(long)n * F_OUT + c] = z;
    XH[(long)n * C_XH + F_IN + c] = h1[(long)n * F_OUT + c] * r;  // XHr columns
  }
}

// ---------------- H-tilde, GRU combine, output head ----------------
__global__ void final_kernel(const float* __restrict__ Ph, const float* __restrict__ QHf,
                             const float* __restrict__ QHf2, const float* __restrict__ QHb,
                             const float* __restrict__ QHb2, const float* __restrict__ bh,
                             const float* __restrict__ Z, const float* __restrict__ h1,
                             const float* __restrict__ Wl, const float* __restrict__ bl,
                             float* __restrict__ y) {
  int n = blockIdx.x * blockDim.x + threadIdx.x;
  if (n >= N_NODES) return;
  float acc = bl[0];
#pragma unroll
  for (int c = 0; c < F_OUT; ++c) {
    float hp = Ph[(long)n * 25 + c] + bh[c] + QHf[(long)n * 10 + c] + 2.0f * QHf2[(long)n * 5 + c] +
               QHb[(long)n * 10 + c] + 2.0f * QHb2[(long)n * 5 + c];
    float ht = tanhf(hp);
    float z = Z[(long)n * F_OUT + c];
    float H = z * h1[(long)n * F_OUT + c] + (1.0f - z) * ht;
    H = H > 0.0f ? H : 0.0f;
    acc += H * Wl[c];
  }
  y[n] = acc;
}

// ---------------- host orchestration ----------------
extern "C" void kernel_launch(void* const* d_in, const int* in_sizes, int n_in,
                              void* d_out, int out_size, void* d_ws, size_t ws_size,
                              hipStream_t stream) {
  const float* x  = (const float*)d_in[0];
  const int*   ei = (const int*)d_in[1];
  const float* ew = (const float*)d_in[2];
  const float* W1 = (const float*)d_in[3];
  const float* b1 = (const float*)d_in[4];
  const float* W2 = (const float*)d_in[5];
  const float* b2 = (const float*)d_in[6];
  const float* Wz = (const float*)d_in[7];
  const float* bz = (const float*)d_in[8];
  const float* Wr = (const float*)d_in[9];
  const float* br = (const float*)d_in[10];
  const float* Wh = (const float*)d_in[11];
  const float* bh = (const float*)d_in[12];
  const float* Wl = (const float*)d_in[13];
  const float* bl = (const float*)d_in[14];
  float* out = (float*)d_out;           // [0,N) = y ; [N, N+E) = A (w_fwd)

  const long N = N_NODES, E = N_EDGES;
  float* Wws = (float*)d_ws;
  float* deg_out = Wws;                 // N
  float* deg_in  = Wws + N;             // N
  float* w_fwd   = Wws + 2 * N;         // E
  float* w_bwd   = w_fwd + E;           // E
  float* h1      = w_bwd + E;           // 5N
  float* Zbuf    = h1 + 5 * N;          // 5N
  float* XH      = Zbuf + 5 * N;        // 40N (becomes XHr in place)
  float* BIG     = XH + 40 * N;         // reuse region, 110N
  float* G1   = BIG;                    // 100N (dead after mlp2)
  float* Proj = BIG;                    // 50N
  float* Qf   = BIG + 50 * N;           // 20N
  float* Qf2  = BIG + 70 * N;           // 10N
  float* Qb   = BIG + 80 * N;           // 20N
  float* Qb2  = BIG + 100 * N;          // 10N
  float* Ph   = BIG;                    // 25N (after Proj is dead)
  float* QHf  = BIG + 25 * N;           // 10N
  float* QHf2 = BIG + 35 * N;           // 5N
  float* QHb  = BIG + 40 * N;           // 10N
  float* QHb2 = BIG + 50 * N;           // 5N
  float* Wc_zr = BIG + 110 * N;         // 40*50
  float* Wc_h  = Wc_zr + C_XH * 50;     // 40*25

  const int* src = ei;                  // edge_index[0]
  const int* dst = ei + E;              // edge_index[1]

  const int B = 256;
  const int gE = (int)((E + B - 1) / B);
  const int gN = (int)((N + B - 1) / B);
  const int gG = (int)((N + 127) / 128);   // 8 waves * 16 rows per block

  // degrees + normalized edge weights
  zero_kernel<<<1024, B, 0, stream>>>(deg_out, 2 * N);
  degree_kernel<<<gE, B, 0, stream>>>(ei, ew, deg_out, deg_in);
  edgew_kernel<<<gE, B, 0, stream>>>(ei, ew, deg_out, deg_in, w_fwd, w_bwd, out + N);

  // MLP warm start: G1 = x@W1 (WMMA), then h1 + XH
  gemm_wmma_kernel<<<gG, B, 0, stream>>>(x, W1, G1, (int)N, F_IN, HID);
  mlp2_kernel<<<gN, B, 0, stream>>>(x, G1, b1, W2, b2, h1, XH);

  // folded projection weights
  buildw_kernel<<<(C_XH * 75 + B - 1) / B, B, 0, stream>>>(Wz, Wr, Wh, Wc_zr, Wc_h);

  // Z/R: project then diffuse (project-first exploits prop(X)@W == prop(X@W))
  gemm_wmma_kernel<<<gG, B, 0, stream>>>(XH, Wc_zr, Proj, (int)N, C_XH, 50);
  zero_kernel<<<1024, B, 0, stream>>>(Qf, 20 * N);
  prop_kernel<<<gE, B, 0, stream>>>(Proj, 50, 10, Qf, 20, src, dst, w_fwd);
  zero_kernel<<<1024, B, 0, stream>>>(Qf2, 10 * N);
  prop_kernel<<<gE, B, 0, stream>>>(Qf, 20, 10, Qf2, 10, src, dst, w_fwd);
  zero_kernel<<<1024, B, 0, stream>>>(Qb, 20 * N);
  prop_kernel<<<gE, B, 0, stream>>>(Proj, 50, 30, Qb, 20, dst, src, w_bwd);
  zero_kernel<<<1024, B, 0, stream>>>(Qb2, 10 * N);
  prop_kernel<<<gE, B, 0, stream>>>(Qb, 20, 10, Qb2, 10, dst, src, w_bwd);
  gates_kernel<<<gN, B, 0, stream>>>(Proj, Qf, Qf2, Qb, Qb2, bz, br, h1, Zbuf, XH);

  // H gate on XHr
  gemm_wmma_kernel<<<gG, B, 0, stream>>>(XH, Wc_h, Ph, (int)N, C_XH, 25);
  zero_kernel<<<1024, B, 0, stream>>>(QHf, 10 * N);
  prop_kernel<<<gE, B, 0, stream>>>(Ph, 25, 5, QHf, 10, src, dst, w_fwd);
  zero_kernel<<<1024, B, 0, stream>>>(QHf2, 5 * N);
  prop_kernel<<<gE, B, 0, stream>>>(QHf, 10, 5, QHf2, 5, src, dst, w_fwd);
  zero_kernel<<<1024, B, 0, stream>>>(QHb, 10 * N);
  prop_kernel<<<gE, B, 0, stream>>>(Ph, 25, 15, QHb, 10, dst, src, w_bwd);
  zero_kernel<<<1024, B, 0, stream>>>(QHb2, 5 * N);
  prop_kernel<<<gE, B, 0, stream>>>(QHb, 10, 5, QHb2, 5, dst, src, w_bwd);

  final_kernel<<<gN, B, 0, stream>>>(Ph, QHf, QHf2, QHb, QHb2, bh, Zbuf, h1, Wl, bl, out);
  (void)in_sizes; (void)n_in; (void)out_size; (void)ws_size;
}